// EmRouting_3796751090153
// MI455X (gfx1250) — compile-verified
//
#include <hip/hip_runtime.h>
#include <math.h>

// ---- problem constants (from reference) ----
#define O_      32          // PARENT_CAPS
#define CH_     16          // POSE
#define KT_     144         // kk * CHILD_CAPS
#define P_      10          // parent_space
#define P2_     100
#define NB_     4
#define CELLS_  400         // NB_*P2_
#define EPSF    1e-9f
#define LNF     (100.0f/4.5f)   // layer_norm_factor = 100/(144/32)
#define TWO_PI_ 6.28318530717958647692f

typedef float v2f __attribute__((ext_vector_type(2)));
typedef float v8f __attribute__((ext_vector_type(8)));

// #parents covering child coordinate x (child_space=12, k=3, stride=1, parent_space=10)
__device__ __forceinline__ int cnt12(int x) {
    int lo = x - 2; if (lo < 0) lo = 0;
    int hi = x;     if (hi > 9) hi = 9;
    return hi - lo + 1;
}

// =====================================================================
// Kernel A: per parent cell — m_step (+ zz log-posterior for the e-step)
// One workgroup (256 thr = 8 waves) per cell. Votes slice (288KB) staged
// into LDS via CDNA5 async global->LDS DMA.
// =====================================================================
__global__ __launch_bounds__(256)
void em_mstep(const float* __restrict__ votes,   // [CELLS][KT][O][CH]
              const float* __restrict__ act_i,   // [CELLS][KT]
              const float* __restrict__ beta_v,  // [O]
              const float* __restrict__ beta_a,  // [O]
              const float* __restrict__ rr_ws,   // [CELLS][KT][O]
              float*       __restrict__ zz_ws,   // [CELLS][KT][O]
              float*       __restrict__ out,     // poses [CELLS][O][CH] ++ acts [CELLS][O]
              float lam, int first, int last)
{
    __shared__ float votes_lds[KT_*O_*CH_];  // 294912 B
    __shared__ float w_s[KT_*O_];            //  18432 B   rr' = rr*act
    __shared__ float s1_s[O_];               // Σ_k rr'
    __shared__ float mean_s[O_*CH_];
    __shared__ float var_s[O_*CH_];
    __shared__ float op2_s[O_];
    __shared__ float lga_s[O_];

    const int cell = blockIdx.x;
    const int tid  = threadIdx.x;
    const int lane = tid & 31;
    const int wv   = tid >> 5;
    const size_t vbase = (size_t)cell * (KT_*O_*CH_);

    // ---- kick off async DMA of the 288KB vote tile into LDS (b128/lane) ----
    {
        const float* g = votes + vbase;
        #pragma unroll 4
        for (int j = 0; j < 72; ++j) {                 // 72*256 lanes * 16B = 288KB
            int e = (j*256 + tid) * 4;                 // float index, 16B aligned
            unsigned lo = (unsigned)(size_t)(&votes_lds[e]);  // low 32b = LDS offset
            const float* ga = g + e;
            asm volatile("global_load_async_to_lds_b128 %0, %1, off"
                         :: "v"(lo), "v"(ga) : "memory");
        }
    }

    // ---- rr' = rr * act_i  (it==0: analytic init rr0 = 1/(ppc*O+eps)) ----
    {
        const int pr = (cell % P2_) / P_, pc = (cell % P2_) % P_;
        for (int idx = tid; idx < KT_*O_; idx += 256) {
            int k = idx >> 5;                          // / O_
            float a = act_i[(size_t)cell*KT_ + k];
            float r;
            if (first) {
                int ks  = k >> 4;                      // kernel slot 0..8
                int row = pr + ks/3, col = pc + ks%3;  // child cell coords
                r = 1.0f / ((float)(cnt12(row)*cnt12(col)) * (float)O_ + EPSF);
            } else {
                r = rr_ws[(size_t)cell*(KT_*O_) + idx];
            }
            w_s[idx] = r * a;
        }
    }
    __syncthreads();

    // ---- rr'_sum per parent cap ----
    if (tid < O_) {
        float s = 0.f;
        for (int k = 0; k < KT_; ++k) s += w_s[k*O_ + tid];
        s1_s[tid] = s;
    }
    asm volatile("s_wait_asynccnt 0" ::: "memory");    // my wave's DMA done
    __syncthreads();                                   // everyone's DMA + s1 visible

    // ---- weighted moments via V_WMMA_F32_16X16X4_F32 ----
    // Per (cell,o): sv[h]=Σ_k w[k]v[k,h], sv2[h]=Σ_k w[k]v[k,h]^2 as
    // D = A(16x4: rows=h, K=4 children) x B(4x16: w replicated across cols).
    // A layout (§7.12.2): lanes0-15 K=0/1 in vgpr0/1, lanes16-31 K=2/3. B mirrored.
    {
        const int h  = lane & 15;
        const int kp = lane >> 4;                      // 0/1 -> K pair {0,1} or {2,3}
        for (int j = 0; j < 4; ++j) {
            const int o = wv*4 + j;                    // 8 waves x 4 = 32 caps
            v8f sv = {0,0,0,0,0,0,0,0}, sv2 = {0,0,0,0,0,0,0,0};
            const float* vl = &votes_lds[o*CH_ + h];
            for (int kb = 0; kb < KT_; kb += 4) {
                int ka = kb + 2*kp;
                float a0 = vl[(ka    )*(O_*CH_)];
                float a1 = vl[(ka + 1)*(O_*CH_)];
                float b0 = w_s[(ka    )*O_ + o];
                float b1 = w_s[(ka + 1)*O_ + o];
                v2f A  = {a0, a1};
                v2f A2 = {a0*a0, a1*a1};
                v2f B  = {b0, b1};
                sv  = __builtin_amdgcn_wmma_f32_16x16x4_f32(false, A,  false, B, (short)0, sv,  false, false);
                sv2 = __builtin_amdgcn_wmma_f32_16x16x4_f32(false, A2, false, B, (short)0, sv2, false, false);
            }
            if (h == 0) {                              // D cols identical; one writer/half
                float s1 = s1_s[o];
                #pragma unroll
                for (int r = 0; r < 8; ++r) {
                    int hh = kp*8 + r;                 // D row m = kp*8+r
                    float sA  = sv[r], sB = sv2[r];
                    float mu  = sA / (s1 + EPSF);
                    float var = (sB - 2.f*mu*sA + mu*mu*s1) / (s1 + EPSF);
                    mean_s[o*CH_ + hh] = mu;
                    var_s [o*CH_ + hh] = var;
                }
            }
        }
    }
    __syncthreads();

    // ---- cost -> activation (wave 0, lane = parent cap), wave32 shfl reduce ----
    if (wv == 0) {
        const int o = lane;
        float s1 = s1_s[o];
        float bv = beta_v[o];
        float cost = 0.f, op2 = 0.f;
        #pragma unroll
        for (int hh = 0; hh < CH_; ++hh) {
            float var = var_s[o*CH_ + hh];
            cost += (bv + 0.5f*logf(var + EPSF)) * s1 * LNF;
            op2  += logf(TWO_PI_ * var);
        }
        op2 *= -0.5f;
        float cm = cost;
        for (int d = 16; d; d >>= 1) cm += __shfl_xor(cm, d, 32);
        cm *= (1.0f/32.0f);
        float dv = cost - cm; dv *= dv;
        for (int d = 16; d; d >>= 1) dv += __shfl_xor(dv, d, 32);
        float sd  = sqrtf(dv * (1.0f/32.0f));
        float act = 1.0f / (1.0f + expf(-(lam * (beta_a[o] + (cm - cost)/(sd + EPSF)))));
        op2_s[o] = op2;
        lga_s[o] = logf(act + EPSF);
        if (last) out[(size_t)(CELLS_*O_*CH_) + (size_t)cell*O_ + o] = act;
    }
    __syncthreads();

    if (last) {   // final iteration: emit poses, no e-step needed
        for (int idx = tid; idx < O_*CH_; idx += 256)
            out[(size_t)cell*(O_*CH_) + idx] = mean_s[idx];
        return;
    }

    // ---- zz = log(act+eps) + o_p0 + o_p2 ; reduce over pose h (16 lanes) ----
    {
        const int h  = tid & 15;
        const int ob = tid >> 4;                       // 0..15
        for (int oo = 0; oo < 2; ++oo) {
            const int o = ob + 16*oo;
            float mu    = mean_s[o*CH_ + h];
            float inv2v = 0.5f / var_s[o*CH_ + h];
            float zb    = lga_s[o] + op2_s[o];
            for (int k = 0; k < KT_; ++k) {
                float d = votes_lds[k*(O_*CH_) + o*CH_ + h] - mu;
                float q = d*d*inv2v;
                for (int m = 8; m; m >>= 1) q += __shfl_xor(q, m, 32); // within 16-lane half
                if (h == 0) zz_ws[(size_t)cell*(KT_*O_) + k*O_ + o] = zb - q;
            }
        }
    }
}

// =====================================================================
// Kernel B: e-step segmented softmax. One wave32 per (n, child cell, child
// cap); lane = parent cap. Gathers <=9 parent (cell,slot) zz rows, softmax
// over (parents x 32 caps), scatters rr.
// =====================================================================
__global__ __launch_bounds__(256)
void em_estep(const float* __restrict__ zz_ws, float* __restrict__ rr_ws)
{
    const int g    = blockIdx.x * 8 + (threadIdx.x >> 5);
    const int lane = threadIdx.x & 31;
    const int i    = g & 15;                  // child cap
    const int c    = (g >> 4) % 144;          // child cell
    const int n    = g / (144*16);
    const int row  = c / 12, col = c % 12;

    float z[9]; int pidx[9]; int cnt = 0;
    for (int dr = 0; dr < 3; ++dr) {
        int pr = row - dr; if (pr < 0 || pr > 9) continue;
        for (int dc = 0; dc < 3; ++dc) {
            int pc = col - dc; if (pc < 0 || pc > 9) continue;
            int p = pr*10 + pc;
            int k = (dr*3 + dc)*16 + i;
            size_t idx = ((size_t)(n*100 + p)*144 + k)*32 + lane;
            z[cnt] = zz_ws[idx];
            pidx[cnt] = (int)idx;
            ++cnt;
        }
    }
    float m = -INFINITY;
    for (int j = 0; j < cnt; ++j) m = fmaxf(m, z[j]);
    for (int d = 16; d; d >>= 1) m = fmaxf(m, __shfl_xor(m, d, 32));
    if (isinf(m)) m = 0.f;
    float s = 0.f;
    for (int j = 0; j < cnt; ++j) s += expf(z[j] - m);
    for (int d = 16; d; d >>= 1) s += __shfl_xor(s, d, 32);
    float inv = 1.0f / s;
    for (int j = 0; j < cnt; ++j) rr_ws[pidx[j]] = expf(z[j] - m) * inv;
}

// =====================================================================
// Host launcher: A(it0,init) -> B -> A(it1) -> B -> A(it2, write out)
// =====================================================================
extern "C" void kernel_launch(void* const* d_in, const int* in_sizes, int n_in,
                              void* d_out, int out_size, void* d_ws, size_t ws_size,
                              hipStream_t stream) {
    (void)in_sizes; (void)n_in; (void)out_size; (void)ws_size;
    const float* votes = (const float*)d_in[0];   // (400,144,32,16)
    const float* acts  = (const float*)d_in[1];   // (400,144,1)
    const float* bv    = (const float*)d_in[2];   // (...,32,1)
    const float* ba    = (const float*)d_in[3];   // (...,32,1)
    // d_in[4] = spatial_routing_matrix (structure derived analytically)
    // d_in[5], d_in[6] = batch_size, kk scalars (compile-time constants here)

    float* rr_ws = (float*)d_ws;                       // 1,843,200 floats
    float* zz_ws = rr_ws + (size_t)CELLS_*KT_*O_;      // 1,843,200 floats
    float* out   = (float*)d_out;                      // 204800 poses + 12800 acts

    for (int it = 0; it < 3; ++it) {
        float lam = 0.01f * (1.0f - powf(0.95f, (float)(it + 1)));
        em_mstep<<<CELLS_, 256, 0, stream>>>(votes, acts, bv, ba, rr_ws, zz_ws, out,
                                             lam, it == 0 ? 1 : 0, it == 2 ? 1 : 0);
        if (it < 2)
            em_estep<<<(NB_*144*16)/8, 256, 0, stream>>>(zz_ws, rr_ws);
    }
}